// VectorQuantizer_84155589198097
// MI455X (gfx1250) — compile-verified
//
#include <hip/hip_runtime.h>
#include <hip/hip_bf16.h>

typedef __attribute__((ext_vector_type(2))) float v2f;
typedef __attribute__((ext_vector_type(8))) float v8f;

#define DIM           256
#define KTILE         16
#define HALF_STEPS    32                  // 32 K-steps (x4 dims) = 128 dims per chain
#define ROWS_PER_WAVE 16
#define WAVES_PB      4
#define ROWS_PB       (ROWS_PER_WAVE * WAVES_PB)   // 64
#define KSPLIT        2                   // codebook halves -> 2x occupancy
#define QROWS_PB      16                  // rows per block in quantize pass

// ---------------- kernel 1: codebook squared norms ----------------
__global__ __launch_bounds__(256) void vq_csq_kernel(const float* __restrict__ cb,
                                                     float* __restrict__ csq, int K) {
  int k = blockIdx.x * blockDim.x + threadIdx.x;
  if (k >= K) return;
  const float4* p = (const float4*)(cb + (size_t)k * DIM);
  float s = 0.0f;
  for (int i = 0; i < DIM / 4; ++i) {
    float4 v = p[i];
    s += v.x * v.x + v.y * v.y + v.z * v.z + v.w * v.w;
  }
  csq[k] = s;
}

// Async DMA of one contiguous 16KB codebook tile into LDS (128 threads x 8 x 16B).
// GLOBAL_LOAD_ASYNC_TO_LDS_B128: copies 16B per lane from global memory to LDS; tracked by ASYNCcnt.
__device__ __forceinline__ void async_stage_tile(const float* __restrict__ src,
                                                 unsigned ldsBase, int tid) {
  const char* g = (const char*)src + tid * 16;
  unsigned l = ldsBase + tid * 16;
#pragma unroll
  for (int i = 0; i < 8; ++i) {
    asm volatile("global_load_async_to_lds_b128 %0, %1, off offset:%2"
                 :: "v"(l), "v"(g), "i"(i * 2048)
                 : "memory");
  }
}

__device__ __forceinline__ void wait_async_all() {
  asm volatile("s_wait_asynccnt 0x0" ::: "memory");
}

// ---------------- kernel 2: WMMA scores + per-split argmin ----------------
// grid.x = (N/64) * KSPLIT ; block = 128 (4 waves, 16 rows each)
__global__ __launch_bounds__(128) void vq_score_kernel(const float* __restrict__ x,
                                                       const float* __restrict__ cb,
                                                       const float* __restrict__ csq,
                                                       float* __restrict__ candV,
                                                       int* __restrict__ candI,
                                                       int K) {
  __shared__ float cbs[2][KTILE * DIM];   // 2 x 16 KB double-buffered codebook tile

  const int split   = blockIdx.x & (KSPLIT - 1);
  const int rb      = blockIdx.x >> 1;            // row-block id
  const int kBase   = split * (K / KSPLIT);
  const int nTiles  = (K / KSPLIT) / KTILE;       // 128

  const int tid  = threadIdx.x;
  const int wave = tid >> 5;
  const int lane = tid & 31;
  const int n16  = lane & 15;
  const int h    = lane >> 4;
  const int rowBase = rb * ROWS_PB;
  const int myRow   = rowBase + wave * ROWS_PER_WAVE + n16;

  // Flat-LDS addressing: low 32 bits of a generic LDS pointer = wave-relative offset.
  const unsigned cbsOff[2] = {(unsigned)(uintptr_t)(&cbs[0][0]),
                              (unsigned)(uintptr_t)(&cbs[1][0])};

  // A panel (16 rows x 256 dims) in VGPRs, pre-scaled by -2.
  // 16x4 f32 A fragment: VGPR0 = K{0,2}(lo/hi half-wave), VGPR1 = K{1,3}.
  v2f a[2 * HALF_STEPS];
  {
    const float* ap = x + (size_t)myRow * DIM;
#pragma unroll
    for (int s = 0; s < 2 * HALF_STEPS; ++s) {
      a[s].x = -2.0f * ap[4 * s + 2 * h];
      a[s].y = -2.0f * ap[4 * s + 2 * h + 1];
    }
  }

  float best[8];
  int   bidx[8];
#pragma unroll
  for (int i = 0; i < 8; ++i) { best[i] = 3.0e38f; bidx[i] = 0; }

  // Prologue: DMA tile 0 into buffer 0.
  async_stage_tile(cb + (size_t)kBase * DIM, cbsOff[0], tid);
  wait_async_all();
  __syncthreads();

  for (int kt = 0; kt < nTiles; ++kt) {
    const int k0 = kBase + kt * KTILE;

    // Prefetch next tile into the alternate buffer; DMA overlaps the WMMA chain.
    if (kt + 1 < nTiles)
      async_stage_tile(cb + (size_t)(k0 + KTILE) * DIM, cbsOff[(kt + 1) & 1], tid);

    // Two independent accumulation chains over D-halves (ILP):
    // c0 covers d 0..127 seeded with ||c||^2 ; c1 covers d 128..255 seeded 0.
    const float cs = csq[k0 + n16];
    v8f c0 = {cs, cs, cs, cs, cs, cs, cs, cs};
    v8f c1 = {0.f, 0.f, 0.f, 0.f, 0.f, 0.f, 0.f, 0.f};

    const float* bp = &cbs[kt & 1][0] + n16 * DIM + 2 * h;
#pragma unroll
    for (int s = 0; s < HALF_STEPS; ++s) {
      v2f b0 = *(const v2f*)(bp + 4 * s);
      v2f b1 = *(const v2f*)(bp + 128 + 4 * s);
      c0 = __builtin_amdgcn_wmma_f32_16x16x4_f32(false, a[s], false, b0,
                                                 (short)0, c0, false, false);
      c1 = __builtin_amdgcn_wmma_f32_16x16x4_f32(false, a[s + HALF_STEPS], false, b1,
                                                 (short)0, c1, false, false);
    }

    const int col = k0 + n16;
#pragma unroll
    for (int i = 0; i < 8; ++i) {
      float sc = c0[i] + c1[i];          // score(row i+8h, col)
      if (sc < best[i]) { best[i] = sc; bidx[i] = col; }
    }

    // Own DMA done, then workgroup barrier makes the next tile visible to all waves.
    wait_async_all();
    __syncthreads();
  }

  // Cross-lane argmin within each 16-lane group (rows 0-7 in lanes 0-15,
  // rows 8-15 in lanes 16-31). Tie-break: smaller index (matches jnp.argmin).
#pragma unroll
  for (int m = 1; m < 16; m <<= 1) {
#pragma unroll
    for (int i = 0; i < 8; ++i) {
      float ov = __shfl_xor(best[i], m, 32);
      int   oi = __shfl_xor(bidx[i], m, 32);
      if (ov < best[i] || (ov == best[i] && oi < bidx[i])) {
        best[i] = ov; bidx[i] = oi;
      }
    }
  }
  if (n16 == 0) {
#pragma unroll
    for (int i = 0; i < 8; ++i) {
      const int row = rowBase + wave * ROWS_PER_WAVE + 8 * h + i;
      candV[(size_t)row * KSPLIT + split] = best[i];
      candI[(size_t)row * KSPLIT + split] = bidx[i];
    }
  }
}

// ---------------- kernel 3: merge splits, quantize, loss partials ----------------
// grid.x = N/16 ; block = 256 (16 threads per row)
__global__ __launch_bounds__(256) void vq_quant_kernel(const float* __restrict__ x,
                                                       const float* __restrict__ cb,
                                                       const float* __restrict__ candV,
                                                       const int* __restrict__ candI,
                                                       float* __restrict__ out,
                                                       float* __restrict__ partials) {
  __shared__ float sm[256];
  const int t      = threadIdx.x;
  const int rloc   = t >> 4;
  const int lane16 = t & 15;
  const int row    = blockIdx.x * QROWS_PB + rloc;

  // Merge the KSPLIT candidates (split 0 holds lower indices -> '<' keeps ties right).
  float v0 = candV[(size_t)row * KSPLIT + 0];
  float v1 = candV[(size_t)row * KSPLIT + 1];
  int   i0 = candI[(size_t)row * KSPLIT + 0];
  int   i1 = candI[(size_t)row * KSPLIT + 1];
  const int idx = (v1 < v0 || (v1 == v0 && i1 < i0)) ? i1 : i0;

  const float* qp = cb + (size_t)idx * DIM;
  const float* xp = x + (size_t)row * DIM;
  float* op = out + (size_t)row * DIM;

  float lsum = 0.0f;
#pragma unroll
  for (int j = 0; j < DIM / 16; ++j) {
    const int d = lane16 + j * 16;
    float q = qp[d], xv = xp[d];
    float df = q - xv;           // stop_gradient(quantized - x)
    op[d] = xv + df;             // straight-through value, reference rounding order
    lsum += df * df;
  }

  sm[t] = lsum;
  __syncthreads();
  for (int st = 128; st > 0; st >>= 1) {
    if (t < st) sm[t] += sm[t + st];
    __syncthreads();
  }
  if (t == 0) partials[blockIdx.x] = sm[0];
}

// ---------------- kernel 4: deterministic loss reduction ----------------
__global__ __launch_bounds__(256) void vq_loss_kernel(const float* __restrict__ partials,
                                                      float* __restrict__ loss_out,
                                                      int n, float scale) {
  __shared__ float sm[256];
  float s = 0.0f;
  for (int i = threadIdx.x; i < n; i += 256) s += partials[i];
  sm[threadIdx.x] = s;
  __syncthreads();
  for (int st = 128; st > 0; st >>= 1) {
    if (threadIdx.x < st) sm[threadIdx.x] += sm[threadIdx.x + st];
    __syncthreads();
  }
  if (threadIdx.x == 0) loss_out[0] = sm[0] * scale;
}

extern "C" void kernel_launch(void* const* d_in, const int* in_sizes, int n_in,
                              void* d_out, int out_size, void* d_ws, size_t ws_size,
                              hipStream_t stream) {
  const float* x  = (const float*)d_in[0];
  const float* cb = (const float*)d_in[1];
  float* out = (float*)d_out;

  const int N = in_sizes[0] / DIM;   // 32768 rows
  const int K = in_sizes[1] / DIM;   // 4096 codes

  // Scratch layout (floats): csq[K] | candV[N*2] | candI[N*2] | partials[N/16]
  float* csq      = (float*)d_ws;
  float* candV    = csq + K;
  int*   candI    = (int*)(candV + (size_t)N * KSPLIT);
  float* partials = (float*)(candI + (size_t)N * KSPLIT);

  vq_csq_kernel<<<(K + 255) / 256, 256, 0, stream>>>(cb, csq, K);

  const int nScoreBlocks = (N / ROWS_PB) * KSPLIT;   // 1024
  vq_score_kernel<<<nScoreBlocks, 128, 0, stream>>>(x, cb, csq, candV, candI, K);

  const int nQuantBlocks = N / QROWS_PB;             // 2048
  vq_quant_kernel<<<nQuantBlocks, 256, 0, stream>>>(x, cb, candV, candI, out, partials);

  vq_loss_kernel<<<1, 256, 0, stream>>>(partials, out + (size_t)N * DIM, nQuantBlocks,
                                        1.25f / ((float)N * (float)DIM));
}